// SetCriterionWeakSup_88751204205210
// MI455X (gfx1250) — compile-verified
//
#include <hip/hip_runtime.h>
#include <math.h>

// Problem constants (from the reference setup)
#define HDIM 192
#define WDIM 192
#define NIMG 64
#define HW (HDIM * WDIM)
#define THRESH 0.3f
#define EPSF 1e-5f

#define ROWS_PER_BLOCK 16
#define TILES_PER_IMG (HDIM / ROWS_PER_BLOCK) // 12
#define HALO 2
#define LROWS (ROWS_PER_BLOCK + 2 * HALO)     // 20

// ---------------------------------------------------------------- utilities

__device__ __forceinline__ float block_sum(float v, float* red, int tid) {
    red[tid] = v;
    __syncthreads();
    #pragma unroll
    for (int s = 128; s > 0; s >>= 1) {
        if (tid < s) red[tid] += red[tid + s];
        __syncthreads();
    }
    float r = red[0];
    __syncthreads();
    return r;
}

__device__ __forceinline__ float sigmoidf_fast(float x) {
    return 1.0f / (1.0f + __expf(-x));
}

// ------------------------------------------------------------- init / final

__global__ void ws_init_kernel(double* acc) {
    if (threadIdx.x < 3) acc[threadIdx.x] = 0.0;
}

__global__ void finalize_kernel(const double* acc, const int* num_masks, float* out) {
    double nm = (double)num_masks[0];
    out[0] = (float)(acc[0] / nm);                       // loss_proj
    double den = acc[2] > 1.0 ? acc[2] : 1.0;            // max(tsim.sum(), 1)
    out[1] = (float)(acc[1] / den / nm);                 // loss_pair
}

// -------------------------------------------------------- projection dice

__global__ __launch_bounds__(256) void proj_kernel(const float* __restrict__ src,
                                                   const float* __restrict__ box,
                                                   double* __restrict__ acc) {
    __shared__ float rms[HDIM], rmb[HDIM];  // row maxima (src_x / box_x)
    __shared__ float cms[WDIM], cmb[WDIM];  // col maxima (src_y / box_y)
    __shared__ float red[256];

    const int tid = threadIdx.x;
    const int n = blockIdx.x;
    const float* s = src + (size_t)n * HW;
    const float* b = box + (size_t)n * HW;

    if (tid < WDIM) {  // column maxima: coalesced across threads
        float ms = -INFINITY, mb = -INFINITY;
        for (int h = 0; h < HDIM; ++h) {
            ms = fmaxf(ms, s[h * WDIM + tid]);
            mb = fmaxf(mb, b[h * WDIM + tid]);
        }
        cms[tid] = ms;
        cmb[tid] = mb;
    }
    if (tid < HDIM) {  // row maxima: each thread scans its own contiguous row
        float ms = -INFINITY, mb = -INFINITY;
        for (int w = 0; w < WDIM; ++w) {
            ms = fmaxf(ms, s[tid * WDIM + w]);
            mb = fmaxf(mb, b[tid * WDIM + w]);
        }
        rms[tid] = ms;
        rmb[tid] = mb;
    }
    __syncthreads();

    float ix = 0.f, ux = 0.f, iy = 0.f, uy = 0.f;
    if (tid < HDIM) {
        float sx = sigmoidf_fast(rms[tid]);
        float bx = rmb[tid];
        ix = sx * bx;
        ux = sx * sx + bx * bx;
        float sy = sigmoidf_fast(cms[tid]);
        float by = cmb[tid];
        iy = sy * by;
        uy = sy * sy + by * by;
    }
    float Ix = block_sum(ix, red, tid);
    float Ux = block_sum(ux, red, tid);
    float Iy = block_sum(iy, red, tid);
    float Uy = block_sum(uy, red, tid);

    if (tid == 0) {
        float lx = 1.0f - 2.0f * Ix / (Ux + EPSF);
        float ly = 1.0f - 2.0f * Iy / (Uy + EPSF);
        atomicAdd(&acc[0], (double)(lx + ly));
    }
}

// ----------------------------------------------------------- pairwise loss

__global__ __launch_bounds__(256) void pair_kernel(const float* __restrict__ src,
                                                   const float* __restrict__ box,
                                                   const float* __restrict__ sims,
                                                   double* __restrict__ acc) {
    __shared__ float raw[LROWS][WDIM];  // staged logits (halo rows)
    __shared__ float lf[LROWS][WDIM];   // log_sigmoid(x)
    __shared__ float lb[LROWS][WDIM];   // log_sigmoid(-x)
    __shared__ float red[256];

    const int tid = threadIdx.x;
    const int n = blockIdx.x / TILES_PER_IMG;
    const int tile = blockIdx.x % TILES_PER_IMG;
    const int row0 = tile * ROWS_PER_BLOCK;
    const float* simg = src + (size_t)n * HW;

    // --- Stage halo tile of logits into LDS with CDNA5 async global->LDS DMA.
    //     OOB rows are never staged; their log values are forced to 0 below,
    //     matching the reference's zero-padding of log_fg/log_bg.
    for (int idx = tid; idx < LROWS * WDIM; idx += 256) {
        int r = idx / WDIM, w = idx - r * WDIM;
        int gh = row0 - HALO + r;
        if (gh >= 0 && gh < HDIM) {
            unsigned lds_addr = (unsigned)(unsigned long long)(uintptr_t)&raw[r][w];
            unsigned long long gaddr =
                (unsigned long long)(uintptr_t)(simg + gh * WDIM + w);
            asm volatile("global_load_async_to_lds_b32 %0, %1, off"
                         :
                         : "v"(lds_addr), "v"(gaddr)
                         : "memory");
        }
    }
    asm volatile("s_wait_asynccnt 0" ::: "memory");
    __syncthreads();

    // --- log_sigmoid pair per staged element (computed once, reused 9x).
    for (int idx = tid; idx < LROWS * WDIM; idx += 256) {
        int r = idx / WDIM, w = idx - r * WDIM;
        int gh = row0 - HALO + r;
        float fl = 0.f, bl = 0.f;
        if (gh >= 0 && gh < HDIM) {
            float x = raw[r][w];
            float l = log1pf(__expf(-fabsf(x)));
            fl = fminf(x, 0.f) - l;   // log_sigmoid(x)
            bl = fminf(-x, 0.f) - l;  // log_sigmoid(-x)
        }
        lf[r][w] = fl;
        lb[r][w] = bl;
    }
    __syncthreads();

    // Neighbor offsets in F.unfold row-major order, k=3 dilation=2, center removed.
    const int dy[8] = {-2, -2, -2, 0, 0, 2, 2, 2};
    const int dx[8] = {-2, 0, 2, -2, 2, -2, 0, 2};

    float num = 0.f, den = 0.f;
    for (int p = tid; p < ROWS_PER_BLOCK * WDIM; p += 256) {
        int hl = p / WDIM, w = p - hl * WDIM;
        int gh = row0 + hl;
        float bx = box[(size_t)n * HW + gh * WDIM + w];
        if (bx != 0.f) {
            float lfc = lf[hl + HALO][w];
            float lbc = lb[hl + HALO][w];
            const float* sp = sims + (size_t)n * 8 * HW + gh * WDIM + w;
            #pragma unroll
            for (int c = 0; c < 8; ++c) {
                float sv = sp[(size_t)c * HW];
                if (sv >= THRESH) {
                    int ww = w + dx[c];
                    float lfn = 0.f, lbn = 0.f;
                    if (ww >= 0 && ww < WDIM) {
                        lfn = lf[hl + HALO + dy[c]][ww];
                        lbn = lb[hl + HALO + dy[c]][ww];
                    }
                    float a = lfc + lfn;
                    float b = lbc + lbn;
                    float m = fmaxf(a, b);
                    float ls = m + log1pf(__expf(-fabsf(a - b)));  // logaddexp
                    num -= ls;  // src_sim = -log_same
                    den += 1.f;
                }
            }
        }
    }

    float bn = block_sum(num, red, tid);
    float bd = block_sum(den, red, tid);
    if (tid == 0) {
        atomicAdd(&acc[1], (double)bn);
        atomicAdd(&acc[2], (double)bd);
    }
}

// --------------------------------------------------------------- launcher

extern "C" void kernel_launch(void* const* d_in, const int* in_sizes, int n_in,
                              void* d_out, int out_size, void* d_ws, size_t ws_size,
                              hipStream_t stream) {
    const float* src  = (const float*)d_in[0];  // (64,192,192)
    const float* box  = (const float*)d_in[1];  // (64,192,192)
    const float* sims = (const float*)d_in[2];  // (64,8,192,192)
    const int*   nm   = (const int*)d_in[3];    // scalar num_masks
    float* out = (float*)d_out;                 // 2 floats
    double* acc = (double*)d_ws;                // [proj_sum, pair_num, pair_den]

    ws_init_kernel<<<1, 32, 0, stream>>>(acc);
    proj_kernel<<<NIMG, 256, 0, stream>>>(src, box, acc);
    pair_kernel<<<NIMG * TILES_PER_IMG, 256, 0, stream>>>(src, box, sims, acc);
    finalize_kernel<<<1, 1, 0, stream>>>(acc, nm, out);
}